// CropbiasLoss_51075751084379
// MI455X (gfx1250) — compile-verified
//
#include <hip/hip_runtime.h>
#include <hip/hip_bf16.h>

// ---------------------------------------------------------------------------
// CropbiasLoss for MI455X (gfx1250, wave32).
// Bandwidth-bound: 285 MB input @ 23.3 TB/s ~= 12 us floor.
// One workgroup per 128x128 map; WMMA f32 16x16x4 used as a wave32 f32
// all-reduce (A = lane partials, B = ones): one v_wmma + one shfl_xor(16)
// reduces 32 lanes at full f32 precision.
// ---------------------------------------------------------------------------

typedef __attribute__((ext_vector_type(2))) float v2f;
typedef __attribute__((ext_vector_type(8))) float v8f;

#define MAP_W   128
#define MAP_N   (128 * 128)
#define CROP    64
#define HALF    32
#define BLOCK   256
#define NWAVE   (BLOCK / 32)

// Full-wave f32 sum via V_WMMA_F32_16X16X4_F32.
// A-layout (ISA 7.12.2): lane m holds A[m][0..1], lane m+16 holds A[m][2..3].
// a = {p, 0} -> row m of A = [p_m, 0, p_{m+16}, 0]; B = ones -> D[m][n] =
// p_m + p_{m+16} for every n. Lane L holds D[v][L%16] (v=0..7; +8 rows for
// lanes 16..31), so sum of the 8 D regs + xor-16 partner = sum of all 32 p.
// EXEC must be all 1s at call sites (full 256-thread blocks, uniform flow).
__device__ __forceinline__ float wave_sum_wmma(float p) {
    v2f a; a.x = p;   a.y = 0.0f;
    v2f b; b.x = 1.0f; b.y = 1.0f;
    v8f c = {};
    c = __builtin_amdgcn_wmma_f32_16x16x4_f32(
        /*neg_a=*/false, a, /*neg_b=*/false, b,
        /*c_mod=*/(short)0, c, /*reuse_a=*/false, /*reuse_b=*/false);
    float s = c[0] + c[1] + c[2] + c[3] + c[4] + c[5] + c[6] + c[7];
    s += __shfl_xor(s, 16, 32);
    return s;  // every lane holds the full 32-lane sum
}

// Wave32 max + first-index-of-max (jnp.argmax tie rule: lowest index).
__device__ __forceinline__ void wave_max_idx(float& v, int& idx) {
    #pragma unroll
    for (int off = 16; off >= 1; off >>= 1) {
        float ov = __shfl_xor(v, off, 32);
        int   oi = __shfl_xor(idx, off, 32);
        if (ov > v || (ov == v && oi < idx)) { v = ov; idx = oi; }
    }
}

__global__ __launch_bounds__(BLOCK)
void crop_loss_kernel(const float* __restrict__ S,
                      const float* __restrict__ T,
                      float* __restrict__ partial) {
    const int map  = blockIdx.x;
    const float* s = S + (size_t)map * MAP_N;
    const float* t = T + (size_t)map * MAP_N;
    const int tid  = threadIdx.x;
    const int lane = tid & 31;
    const int wav  = tid >> 5;

    __shared__ float sh_a[NWAVE];
    __shared__ float sh_b[NWAVE];
    __shared__ float sh_m[NWAVE];
    __shared__ int   sh_i[NWAVE];
    __shared__ float bc_cs, bc_ct;
    __shared__ int   bc_tx, bc_ty;

    // ---------------- Phase 1: stream both maps (coalesced float4) ---------
    float cs = 0.0f, ct = 0.0f;
    float tmax = -__builtin_inff();
    int   tidx = 0x7fffffff;
    #pragma unroll 4
    for (int i = 0; i < MAP_N / (BLOCK * 4); ++i) {
        const int base = (i * BLOCK + tid) * 4;          // ascending per thread
        const float4 sv = *(const float4*)(s + base);
        const float4 tv = *(const float4*)(t + base);
        cs += __expf(sv.x) + __expf(sv.y) + __expf(sv.z) + __expf(sv.w);
        ct += __expf(tv.x) + __expf(tv.y) + __expf(tv.z) + __expf(tv.w);
        const float tvv[4] = {tv.x, tv.y, tv.z, tv.w};
        #pragma unroll
        for (int j = 0; j < 4; ++j)
            if (tvv[j] > tmax) { tmax = tvv[j]; tidx = base + j; }  // first-occurrence
    }
    cs = wave_sum_wmma(cs);
    ct = wave_sum_wmma(ct);
    wave_max_idx(tmax, tidx);
    if (lane == 0) { sh_a[wav] = cs; sh_b[wav] = ct; sh_m[wav] = tmax; sh_i[wav] = tidx; }
    __syncthreads();
    if (tid == 0) {
        float a = 0.0f, b = 0.0f, m = -__builtin_inff(); int mi = 0x7fffffff;
        #pragma unroll
        for (int w = 0; w < NWAVE; ++w) {
            a += sh_a[w]; b += sh_b[w];
            if (sh_m[w] > m || (sh_m[w] == m && sh_i[w] < mi)) { m = sh_m[w]; mi = sh_i[w]; }
        }
        bc_cs = a; bc_ct = b; bc_ty = mi / MAP_W; bc_tx = mi % MAP_W;
    }
    __syncthreads();

    // ---------------- Phase 2: crop coordinates (f32, as reference) --------
    const float csf = bc_cs, ctf = bc_ct;
    const int tx = bc_tx, ty = bc_ty;
    const float scale = csf / (csf - 1.0f);
    const float bias  = 1.0f / (2.0f * csf);
    const int sx = (int)truncf(scale * ((float)tx - bias));
    const int sy = (int)truncf(scale * ((float)ty - bias));
    // mirror pads use the UNclamped start; dynamic_slice clamps to [0,128]
    const int s_pt = max(0, HALF - sy), s_pb = max(0, sy - (MAP_W - HALF));
    const int s_pl = max(0, HALF - sx), s_pr = max(0, sx - (MAP_W - HALF));
    const int t_pt = max(0, HALF - ty), t_pb = max(0, ty - (MAP_W - HALF));
    const int t_pl = max(0, HALF - tx), t_pr = max(0, tx - (MAP_W - HALF));
    const int syc = min(max(sy, 0), MAP_W), sxc = min(max(sx, 0), MAP_W);
    const int tyc = min(max(ty, 0), MAP_W), txc = min(max(tx, 0), MAP_W);
    const float inv_cs = 1.0f / csf, inv_ct = 1.0f / ctf;

    // ---------------- Phase 3: crop gather + diff^2 (hits L2) --------------
    float acc = 0.0f;
    #pragma unroll 4
    for (int e = tid; e < CROP * CROP; e += BLOCK) {
        const int r = e >> 6, c = e & 63;
        const int rs = (r < s_pt || r >= CROP - s_pb) ? (CROP - 1 - r) : r;
        const int cs2 = (c < s_pl || c >= CROP - s_pr) ? (CROP - 1 - c) : c;
        const int rt = (r < t_pt || r >= CROP - t_pb) ? (CROP - 1 - r) : r;
        const int ct2 = (c < t_pl || c >= CROP - t_pr) ? (CROP - 1 - c) : c;
        const int ys = syc + rs - HALF, xs = sxc + cs2 - HALF;
        const int yt = tyc + rt - HALF, xt = txc + ct2 - HALF;
        float sval = 0.0f, tval = 0.0f;
        if ((unsigned)ys < (unsigned)MAP_W && (unsigned)xs < (unsigned)MAP_W)
            sval = __expf(s[ys * MAP_W + xs]) * inv_cs;   // softmax = exp/sum(exp)
        if ((unsigned)yt < (unsigned)MAP_W && (unsigned)xt < (unsigned)MAP_W)
            tval = __expf(t[yt * MAP_W + xt]) * inv_ct;
        const float d = sval - tval;
        acc += d * d;
    }
    acc = wave_sum_wmma(acc);            // uniform flow: EXEC all-1s here
    if (lane == 0) sh_a[wav] = acc;
    __syncthreads();
    if (tid == 0) {
        float tot = 0.0f;
        #pragma unroll
        for (int w = 0; w < NWAVE; ++w) tot += sh_a[w];
        partial[map] = tot;
    }
}

// Deterministic 2176 -> 1 reduction (single block), writes total / B.
__global__ __launch_bounds__(BLOCK)
void final_reduce_kernel(const float* __restrict__ partial, int n,
                         float* __restrict__ out, float inv_b) {
    const int tid = threadIdx.x;
    const int lane = tid & 31;
    const int wav  = tid >> 5;
    __shared__ float sh[NWAVE];
    float acc = 0.0f;
    for (int i = tid; i < n; i += BLOCK) acc += partial[i];
    acc = wave_sum_wmma(acc);
    if (lane == 0) sh[wav] = acc;
    __syncthreads();
    if (tid == 0) {
        float tot = 0.0f;
        #pragma unroll
        for (int w = 0; w < NWAVE; ++w) tot += sh[w];
        out[0] = tot * inv_b;
    }
}

extern "C" void kernel_launch(void* const* d_in, const int* in_sizes, int n_in,
                              void* d_out, int out_size, void* d_ws, size_t ws_size,
                              hipStream_t stream) {
    const float* S = (const float*)d_in[0];   // s_feature [64,34,128,128] f32
    const float* T = (const float*)d_in[1];   // t_feature [64,34,128,128] f32
    float* out     = (float*)d_out;           // scalar f32
    float* partial = (float*)d_ws;            // one f32 per map

    const int nmaps = in_sizes[0] / MAP_N;    // 64*34 = 2176
    crop_loss_kernel<<<nmaps, BLOCK, 0, stream>>>(S, T, partial);
    final_reduce_kernel<<<1, BLOCK, 0, stream>>>(partial, nmaps, out, 1.0f / 64.0f);
}